// psf_model_34351148434250
// MI455X (gfx1250) — compile-verified
//
#include <hip/hip_runtime.h>

typedef __attribute__((ext_vector_type(2))) float v2f;
typedef __attribute__((ext_vector_type(8))) float v8f;

#define T_DIM   400
#define Z_DIM   128
#define NCOL    256                 // 2*Z (re/im interleaved columns)
#define M_DIM   16384               // 128*128
#define KC      40                  // K chunk (divides 400, multiple of 4)
#define LDA     42                  // padded LDS strides (even, conflict-free mod 64 banks)
#define LDB     42
#define OUTC    ((size_t)128*128*128*2)   // floats per complex component
#define A_CONST 5.9052493f          // pi / 0.532

// ---------------------------------------------------------------------------
// Kernel 1: build B[t, 2z]=w[t]*cos(ab[t]-U[z]*cos(th)), B[t,2z+1]=w[t]*sin(.)
// ---------------------------------------------------------------------------
__global__ __launch_bounds__(256) void build_kern(const float* __restrict__ ab,
                                                  const float* __restrict__ theta,
                                                  const float* __restrict__ U,
                                                  float* __restrict__ Bg) {
    int i = blockIdx.x * 256 + threadIdx.x;
    if (i >= T_DIM * Z_DIM) return;
    int t = i >> 7, z = i & 127;
    float th = theta[t];
    float dp = (t > 0)         ? (th - theta[t - 1]) : 0.f;
    float dn = (t < T_DIM - 1) ? (theta[t + 1] - th) : 0.f;
    float w  = 0.5f * (dp + dn);                 // trapz weight
    float ph = ab[t] - U[z] * cosf(th);
    float s, c;
    sincosf(ph, &s, &c);
    Bg[t * NCOL + 2 * z]     = w * c;
    Bg[t * NCOL + 2 * z + 1] = w * s;
}

// ---------------------------------------------------------------------------
// Kernel 2: fused 3x (16384x400 @ 400x256) fp32 WMMA GEMM + Phi epilogue
// ---------------------------------------------------------------------------
__global__ __launch_bounds__(256) void psf_wmma(const float* __restrict__ F0,
                                                const float* __restrict__ F1,
                                                const float* __restrict__ F2,
                                                const float* __restrict__ Phi,
                                                const float* __restrict__ Bg,
                                                float* __restrict__ out) {
    __shared__ float ldsA[3 * 32 * LDA];   // 16128 B
    __shared__ float ldsB[NCOL * LDB];     // 43008 B

    const int tid  = threadIdx.x;
    const int lane = tid & 31;
    const int wave = tid >> 5;
    const int wm   = wave >> 2;            // 0..1 : which 16-row half
    const int wn   = wave & 3;             // 0..3 : which 64-col strip
    const int r0   = blockIdx.x * 32;      // block's first M row

    const int koff = (lane >> 4) << 1;     // lanes 16-31 hold K+2,K+3
    const int l16  = lane & 15;

    v8f acc0[4], acc1[4], acc2[4];
    const v8f vzero = {0.f, 0.f, 0.f, 0.f, 0.f, 0.f, 0.f, 0.f};
#pragma unroll
    for (int t = 0; t < 4; ++t) { acc0[t] = vzero; acc1[t] = vzero; acc2[t] = vzero; }

    for (int kc = 0; kc < T_DIM; kc += KC) {
        // ---- stage A: 3 comps x 32 rows x KC (row-contiguous global reads) ----
        for (int idx = tid; idx < 32 * KC; idx += 256) {
            int row = idx / KC, k = idx % KC;
            int g = (r0 + row) * T_DIM + kc + k;
            int l = row * LDA + k;
            ldsA[0 * 32 * LDA + l] = F0[g];
            ldsA[1 * 32 * LDA + l] = F1[g];
            ldsA[2 * 32 * LDA + l] = F2[g];
        }
        // ---- stage B transposed: ldsB[n*LDB + k] (coalesced reads, pad-42 writes) ----
        for (int idx = tid; idx < KC * NCOL; idx += 256) {
            int k = idx >> 8, n = idx & 255;
            ldsB[n * LDB + k] = Bg[(kc + k) * NCOL + n];
        }
        if (kc + KC < T_DIM) {   // warm next chunk (global_prefetch_b8)
            __builtin_prefetch(&Bg[(size_t)(kc + KC) * NCOL + tid], 0, 0);
            __builtin_prefetch(&F0[(size_t)(r0 + (tid >> 3)) * T_DIM + kc + KC], 0, 0);
        }
        __syncthreads();

        // A fragment base for this lane: row = wm*16 + l16, K pair at koff
        const float* aBase = &ldsA[(wm * 16 + l16) * LDA + koff];
#pragma unroll
        for (int k0 = 0; k0 < KC; k0 += 4) {
            v2f a0 = *(const v2f*)&aBase[0 * 32 * LDA + k0];
            v2f a1 = *(const v2f*)&aBase[1 * 32 * LDA + k0];
            v2f a2 = *(const v2f*)&aBase[2 * 32 * LDA + k0];
#pragma unroll
            for (int t = 0; t < 4; ++t) {
                int n = wn * 64 + t * 16 + l16;
                v2f b = *(const v2f*)&ldsB[n * LDB + k0 + koff];
                acc0[t] = __builtin_amdgcn_wmma_f32_16x16x4_f32(
                    false, a0, false, b, (short)0, acc0[t], false, false);
                acc1[t] = __builtin_amdgcn_wmma_f32_16x16x4_f32(
                    false, a1, false, b, (short)0, acc1[t], false, false);
                acc2[t] = __builtin_amdgcn_wmma_f32_16x16x4_f32(
                    false, a2, false, b, (short)0, acc2[t], false, false);
            }
        }
        __syncthreads();
    }

    // ---- epilogue: Ex = i*A*(I0+I2*cos2p), Ey = i*A*I2*sin2p, Ez = -2A*I1*cosp
    // C layout: lane l, vgpr j -> M = 8*(l>>4)+j, N = l&15.
    // Column n = 2z+p: p==0 holds Re(I), p==1 holds Im(I).
    const int gm_base = r0 + wm * 16 + ((lane >> 4) << 3);
#pragma unroll
    for (int j = 0; j < 8; ++j) {
        int m = gm_base + j;
        float phi = Phi[m];
        float s2, c2;
        sincosf(2.f * phi, &s2, &c2);
        float cp = cosf(phi);
        size_t ro = (size_t)m * NCOL;
#pragma unroll
        for (int t = 0; t < 4; ++t) {
            int n = wn * 64 + t * 16 + l16;
            float i0 = acc0[t][j], i1 = acc1[t][j], i2 = acc2[t][j];
            float vx = i0 + i2 * c2;   // component of (I0 + I2*cos2p)
            float vy = i2 * s2;        // component of I2*sin2p
            float vz = i1 * cp;        // component of I1*cosp
            int   p    = n & 1;
            int   offn = p ? (n - 1) : (n + 1);      // i* swaps re<->im slots
            float sg   = p ? -A_CONST : A_CONST;     // i*(R+iI) = -I + iR
            out[ro + offn]             = sg * vx;            // Ex
            out[OUTC + ro + offn]      = sg * vy;            // Ey
            out[2 * OUTC + ro + n]     = -2.f * A_CONST * vz; // Ez (real scale)
        }
    }
}

// ---------------------------------------------------------------------------
extern "C" void kernel_launch(void* const* d_in, const int* in_sizes, int n_in,
                              void* d_out, int out_size, void* d_ws, size_t ws_size,
                              hipStream_t stream) {
    (void)in_sizes; (void)n_in; (void)out_size; (void)ws_size;
    const float* ab    = (const float*)d_in[0];
    const float* F0    = (const float*)d_in[1];
    const float* F1    = (const float*)d_in[2];
    const float* F2    = (const float*)d_in[3];
    const float* Phi   = (const float*)d_in[4];
    const float* theta = (const float*)d_in[5];
    const float* U     = (const float*)d_in[6];
    float* Bg  = (float*)d_ws;                 // 400*256*4 = 409600 B
    float* out = (float*)d_out;

    build_kern<<<(T_DIM * Z_DIM + 255) / 256, 256, 0, stream>>>(ab, theta, U, Bg);
    psf_wmma<<<M_DIM / 32, 256, 0, stream>>>(F0, F1, F2, Phi, Bg, out);
}